// MLLoss_sim_53712861004461
// MI455X (gfx1250) — compile-verified
//
#include <hip/hip_runtime.h>

typedef __attribute__((ext_vector_type(2))) float v2f;
typedef __attribute__((ext_vector_type(4))) float v4f;
typedef __attribute__((ext_vector_type(8))) float v8f;

#define THREADS 256
#define BLOCKS  1024
// products: 32768 rows * 1024 cols = 33554432 elements
#define NPROD   33554432
// per grid pass: BLOCKS*THREADS*4 = 1048576 -> exactly 32 uniform iterations

__global__ __launch_bounds__(THREADS)
void dot_wmma_kernel(const float* __restrict__ real,
                     const float* __restrict__ fake,
                     float* __restrict__ partials) {
    const int stride = BLOCKS * THREADS * 4;
    int idx = (blockIdx.x * THREADS + threadIdx.x) * 4;

    v2f ones; ones[0] = 1.0f; ones[1] = 1.0f;
    v8f c0 = {};
    v8f c1 = {};

    // Uniform trip count (NPROD % stride == 0) -> no divergence, EXEC all-1s
    // as required by WMMA.
    for (int i = idx; i < NPROD; i += stride) {
        v4f r = __builtin_nontemporal_load((const v4f*)(real + i));
        v4f f = __builtin_nontemporal_load((const v4f*)(fake + i));
        v2f a0; a0[0] = r[0] * f[0]; a0[1] = r[1] * f[1];
        v2f a1; a1[0] = r[2] * f[2]; a1[1] = r[3] * f[3];
        // D = A x B + C with B == ones: every product contributes to exactly
        // 16 entries of D regardless of operand striping -> layout-agnostic.
        c0 = __builtin_amdgcn_wmma_f32_16x16x4_f32(
                 false, a0, false, ones, (short)0, c0, false, false);
        c1 = __builtin_amdgcn_wmma_f32_16x16x4_f32(
                 false, a1, false, ones, (short)0, c1, false, false);
    }

    v8f c = c0 + c1;
    float s = c[0] + c[1] + c[2] + c[3] + c[4] + c[5] + c[6] + c[7];

    // wave32 reduction
    for (int off = 16; off > 0; off >>= 1)
        s += __shfl_xor(s, off, 32);

    __shared__ float lsum[THREADS / 32];
    const int lane = threadIdx.x & 31;
    const int wave = threadIdx.x >> 5;
    if (lane == 0) lsum[wave] = s;
    __syncthreads();
    if (threadIdx.x == 0) {
        float b = 0.0f;
        #pragma unroll
        for (int w = 0; w < THREADS / 32; ++w) b += lsum[w];
        partials[blockIdx.x] = b;   // deterministic: one slot per block
    }
}

__global__ __launch_bounds__(256)
void reduce_kernel(const float* __restrict__ partials, float* __restrict__ out) {
    __shared__ float sm[256];
    float s = 0.0f;
    for (int i = threadIdx.x; i < BLOCKS; i += 256)
        s += partials[i];
    sm[threadIdx.x] = s;
    __syncthreads();
    for (int step = 128; step > 0; step >>= 1) {
        if (threadIdx.x < step) sm[threadIdx.x] += sm[threadIdx.x + step];
        __syncthreads();
    }
    if (threadIdx.x == 0) {
        // accumulated = 16 * S  (WMMA column replication)
        // loss = S / 65536 + 0.5 = accumulated / 1048576 + 0.5
        out[0] = sm[0] * (1.0f / 1048576.0f) + 0.5f;
    }
}

extern "C" void kernel_launch(void* const* d_in, const int* in_sizes, int n_in,
                              void* d_out, int out_size, void* d_ws, size_t ws_size,
                              hipStream_t stream) {
    const float* x    = (const float*)d_in[0];
    const float* real = x;                    // rows [0, 32768)
    const float* fake = x + (size_t)NPROD;    // rows [32768, 65536)
    float* partials   = (float*)d_ws;         // BLOCKS floats = 4 KB scratch
    float* out        = (float*)d_out;

    dot_wmma_kernel<<<BLOCKS, THREADS, 0, stream>>>(real, fake, partials);
    reduce_kernel<<<1, 256, 0, stream>>>(partials, out);
}